// DSDM_70351564308696
// MI455X (gfx1250) — compile-verified
//
#include <hip/hip_runtime.h>
#include <hip/hip_bf16.h>

// ---------------------------------------------------------------------------
// DSDM memory update, MI455X (gfx1250), wave32.
// Pass 1: dist[i] = ||A[i]-qa|| via V_WMMA_F32_16X16X4_F32 (diff x diff^T diag),
//         plus global min via atomicMin on float bits.
// Pass 2: sumexp = sum(exp(min_d - dist[i])) via block reduce + atomicAdd.
// Pass 3: one thread per output element, out = append ? v : v + w*(q - v).
// Bandwidth-bound: ~1.2 GB total traffic -> ~52 us floor at 23.3 TB/s.
// ---------------------------------------------------------------------------

typedef __attribute__((ext_vector_type(2))) float v2f;
typedef __attribute__((ext_vector_type(8))) float v8f;

#define D_ADDR    64
#define N_CLASS   10
#define ROW_OUT   (D_ADDR + N_CLASS)   // 74
#define TILE_ROWS 128                  // rows per workgroup in pass 1
#define LDS_PAD   68                   // row stride in floats: conflict-free ds_load_b64

__device__ __forceinline__ float pick8(v8f c, int i) {
    float r = c[0];
    r = (i == 1) ? c[1] : r;
    r = (i == 2) ? c[2] : r;
    r = (i == 3) ? c[3] : r;
    r = (i == 4) ? c[4] : r;
    r = (i == 5) ? c[5] : r;
    r = (i == 6) ? c[6] : r;
    r = (i == 7) ? c[7] : r;
    return r;
}

__global__ void dsdm_init_kernel(unsigned* minBits, float* sumExp) {
    *minBits = 0x7F800000u;   // +inf bit pattern
    *sumExp  = 0.0f;
}

// ---------------- Pass 1: distances + global min --------------------------
__global__ __launch_bounds__(256) void dsdm_dist_kernel(
    const float* __restrict__ A, const float* __restrict__ qa,
    float* __restrict__ dist, unsigned* __restrict__ minBits, int N)
{
    __shared__ float ldsA[TILE_ROWS * LDS_PAD];  // ~34 KB
    __shared__ float ldsQ[D_ADDR];
    __shared__ float red[256];

    const int t        = threadIdx.x;
    const int tileBase = blockIdx.x * TILE_ROWS;

    if (t < D_ADDR) ldsQ[t] = qa[t];

    // Stage 128x64 tile, coalesced global_load_b128, padded rows in LDS.
    #pragma unroll
    for (int i = 0; i < (TILE_ROWS * D_ADDR) / (256 * 4); ++i) {  // 8 iters
        int e   = (t + i * 256) * 4;
        int row = e >> 6;         // /64
        int col = e & 63;
        float4 v = make_float4(0.f, 0.f, 0.f, 0.f);
        if (tileBase + row < N)
            v = *(const float4*)(A + (long)(tileBase + row) * D_ADDR + col);
        *(float4*)(ldsA + row * LDS_PAD + col) = v;
    }
    __syncthreads();

    // Each wave handles 16 rows. A-operand 16x4 f32 layout:
    // lanes 0-15: (row r, K=k,k+1), lanes 16-31: (row r, K=k+2,k+3).
    // Feeding the same regs as B gives D = diff * diff^T; diag = row ssq.
    const int wave = t >> 5;
    const int lane = t & 31;
    const int half = lane >> 4;
    const int r    = lane & 15;
    const float* rowPtr = ldsA + (wave * 16 + r) * LDS_PAD;

    v8f c = {};
    #pragma unroll
    for (int kc = 0; kc < 16; ++kc) {              // K = 64 in chunks of 4
        int k = kc * 4 + half * 2;
        v2f a;
        a.x = rowPtr[k]     - ldsQ[k];
        a.y = rowPtr[k + 1] - ldsQ[k + 1];
        c = __builtin_amdgcn_wmma_f32_16x16x4_f32(
                false, a, false, a, (short)0, c, false, false);
    }

    // Diagonal extraction: row m (m<8) -> lane m, c[m];
    //                      row m (m>=8) -> lane m+16, c[m-8].
    const bool holds = (lane < 8) || (lane >= 24);
    const int  diagRow = (lane < 8) ? lane : (lane - 16);
    const int  vidx    = (lane < 8) ? lane : (lane - 24);
    const float ssq    = pick8(c, vidx);

    float myDist = __builtin_inff();
    const int grow = tileBase + wave * 16 + diagRow;
    if (holds && grow < N) {
        float dd = sqrtf(ssq);
        dist[grow] = dd;
        myDist = dd;
    }

    red[t] = myDist;
    __syncthreads();
    for (int s = 128; s > 0; s >>= 1) {
        if (t < s) red[t] = fminf(red[t], red[t + s]);
        __syncthreads();
    }
    if (t == 0) atomicMin(minBits, __float_as_uint(red[0]));  // dists >= 0
}

// ---------------- Pass 2: sum of exp(min_d - dist) ------------------------
__global__ __launch_bounds__(256) void dsdm_sumexp_kernel(
    const float* __restrict__ dist, const unsigned* __restrict__ minBits,
    float* __restrict__ sumExp, int N)
{
    __shared__ float red[256];
    const float md = __uint_as_float(*minBits);
    float acc = 0.0f;
    for (int i = blockIdx.x * 256 + threadIdx.x; i < N; i += gridDim.x * 256)
        acc += __expf(md - dist[i]);
    red[threadIdx.x] = acc;
    __syncthreads();
    for (int s = 128; s > 0; s >>= 1) {
        if (threadIdx.x < s) red[threadIdx.x] += red[threadIdx.x + s];
        __syncthreads();
    }
    if (threadIdx.x == 0) atomicAdd(sumExp, red[0]);
}

// ---------------- Pass 3: fused update + concat write ---------------------
__global__ __launch_bounds__(256) void dsdm_out_kernel(
    const float* __restrict__ A, const float* __restrict__ M,
    const float* __restrict__ qa, const float* __restrict__ qc,
    const float* __restrict__ geIn, const float* __restrict__ dist,
    const unsigned* __restrict__ minBits, const float* __restrict__ sumExp,
    float* __restrict__ out, int N)
{
    const int idx = blockIdx.x * 256 + threadIdx.x;
    if (idx >= N * ROW_OUT) return;
    const int row = idx / ROW_OUT;
    const int col = idx - row * ROW_OUT;

    const float md  = __uint_as_float(*minBits);
    const float sum = *sumExp;
    const float ge  = geIn[0];

    const float ema      = 2.0f / 101.0f;          // 2/(TIME_PERIOD+1)
    const float ema_temp = 2.0f / (ema + 1.0f);    // 2/(ema+1)
    const float new_ge   = ge + ema_temp * (md - ge);
    const bool  append   = md >= new_ge * 0.95f;   // COEF_GLOBAL_ERROR

    const float w = __expf(md - dist[row]) / sum * ema;

    float v, q;
    if (col < D_ADDR) { v = A[(long)row * D_ADDR + col];          q = qa[col]; }
    else              { v = M[(long)row * N_CLASS + (col - D_ADDR)]; q = qc[col - D_ADDR]; }

    out[idx] = append ? v : fmaf(w, q - v, v);
}

extern "C" void kernel_launch(void* const* d_in, const int* in_sizes, int n_in,
                              void* d_out, int out_size, void* d_ws, size_t ws_size,
                              hipStream_t stream) {
    const float* A  = (const float*)d_in[0];
    const float* M  = (const float*)d_in[1];
    const float* qa = (const float*)d_in[2];
    const float* qc = (const float*)d_in[3];
    const float* ge = (const float*)d_in[4];
    float* out = (float*)d_out;

    const int N = in_sizes[0] / D_ADDR;   // 2,000,000

    // Workspace layout: [0] min bits (u32), [1] sumexp (f32), [16..16+N) dist.
    unsigned* minBits = (unsigned*)d_ws;
    float*    sumExp  = (float*)d_ws + 1;
    float*    dist    = (float*)d_ws + 16;

    dsdm_init_kernel<<<1, 1, 0, stream>>>(minBits, sumExp);

    const int distBlocks = (N + TILE_ROWS - 1) / TILE_ROWS;   // 15625
    dsdm_dist_kernel<<<distBlocks, 256, 0, stream>>>(A, qa, dist, minBits, N);

    dsdm_sumexp_kernel<<<2048, 256, 0, stream>>>(dist, minBits, sumExp, N);

    const long long total = (long long)N * ROW_OUT;           // 148M
    const int outBlocks = (int)((total + 255) / 256);
    dsdm_out_kernel<<<outBlocks, 256, 0, stream>>>(
        A, M, qa, qc, ge, dist, minBits, sumExp, out, N);
}